// DenseCaptioningLoss_87643102642642
// MI455X (gfx1250) — compile-verified
//
#include <hip/hip_runtime.h>
#include <hip/hip_bf16.h>

// Problem constants from the reference
#define BSZ 16
#define MC  16
#define TC  32
#define VC  12000
#define PC  64
#define PVC 20
#define NROWS (BSZ * MC * TC)   // 8192 caption rows
#define NPROG (BSZ * PC)        // 1024 program rows

typedef __attribute__((ext_vector_type(2))) float v2f;
typedef __attribute__((ext_vector_type(4))) float f4;
typedef __attribute__((ext_vector_type(8))) float v8f;

__device__ __forceinline__ float warp_xor_f(float v, int mask) {
    return __shfl_xor(v, mask, 32);
}

// Hardware transcendentals: v_exp_f32 (2^x) and v_log_f32 (log2 x)
__device__ __forceinline__ float hw_exp2(float x) { return __builtin_amdgcn_exp2f(x); }
__device__ __forceinline__ float hw_log2(float x) { return __builtin_amdgcn_logf(x); }

// ---------------------------------------------------------------------------
// Kernel A: one workgroup (8 waves) per (b, m, t) caption row.
// Fused log-softmax + NLL, single memory pass (online softmax in log2 domain),
// block-uniform early exit for masked rows, non-temporal b128 streaming loads.
// Writes per-row nll and valid-count to workspace (deterministic, no atomics).
// ---------------------------------------------------------------------------
__global__ __launch_bounds__(256)
void cap_ce_rows_kernel(const float* __restrict__ pred_captions,
                        const int*   __restrict__ gt_captions,
                        const int*   __restrict__ gt_cap_lens,
                        const int*   __restrict__ gt_caps_count,
                        float* __restrict__ row_nll,
                        float* __restrict__ row_cnt) {
    const int r    = blockIdx.x;          // 0..8191
    const int b    = r >> 9;              // / (MC*TC)
    const int rem  = r & 511;
    const int mcap = rem >> 5;            // / TC
    const int t    = rem & 31;

    // Block-uniform mask: skip the 48KB row entirely when masked out.
    const bool valid = (t < gt_cap_lens[b * MC + mcap]) &&
                       (mcap < gt_caps_count[b]);
    if (!valid) {
        if (threadIdx.x == 0) { row_nll[r] = 0.0f; row_cnt[r] = 0.0f; }
        return;
    }

    const float LOG2E = 1.4426950408889634f;
    const float LN2   = 0.6931471805599453f;

    const float* __restrict__ row = pred_captions + (size_t)r * VC;
    const f4* __restrict__ row4   = (const f4*)row;

    // Online softmax in log2 domain: track max(y), s = sum 2^(y - max)
    float mmax = -3.0e38f;
    float s    = 0.0f;
    for (int i = threadIdx.x; i < VC / 4; i += 256) {
        f4 x = __builtin_nontemporal_load(row4 + i);   // streamed once: NT load
#pragma unroll
        for (int j = 0; j < 4; ++j) {
            float y = x[j] * LOG2E;
            if (y > mmax) {                 // rescale only on new max (rare)
                s = s * hw_exp2(mmax - y);
                mmax = y;
            }
            s += hw_exp2(y - mmax);
        }
    }

    // wave32 butterfly combine of (max, sum) states via ds-permute shuffles
#pragma unroll
    for (int off = 16; off >= 1; off >>= 1) {
        float om = warp_xor_f(mmax, off);
        float os = warp_xor_f(s, off);
        float mn = fmaxf(mmax, om);
        s = s * hw_exp2(mmax - mn) + os * hw_exp2(om - mn);
        mmax = mn;
    }

    // cross-wave combine (8 waves), fixed order -> deterministic
    __shared__ float sm[8], ss[8];
    const int wave = threadIdx.x >> 5;
    const int lane = threadIdx.x & 31;
    if (lane == 0) { sm[wave] = mmax; ss[wave] = s; }
    __syncthreads();

    if (threadIdx.x == 0) {
        float M0 = sm[0], S0 = ss[0];
#pragma unroll
        for (int w = 1; w < 8; ++w) {
            float mn = fmaxf(M0, sm[w]);
            S0 = S0 * hw_exp2(M0 - mn) + ss[w] * hw_exp2(sm[w] - mn);
            M0 = mn;
        }
        const int gt = gt_captions[r];                 // same flat [BS,M,T] index
        const float xg = row[gt];
        // lse = (max + log2(sum)) * ln2 ; nll = lse - logit[gt]
        row_nll[r] = (M0 + hw_log2(S0)) * LN2 - xg;
        row_cnt[r] = 1.0f;
    }
}

// ---------------------------------------------------------------------------
// Kernel B: single workgroup. Program CE (1024 x 20), IoU (256), plus the
// deterministic final reduction of all six scalars. The 256-partials-per-
// quantity reduction is done with chained V_WMMA_F32_16X16X4_F32 against an
// all-ones B matrix: D = A*1 + C collapses each 16x4 tile into row sums.
// ---------------------------------------------------------------------------
__global__ __launch_bounds__(256)
void finalize_kernel(const float* __restrict__ row_nll,
                     const float* __restrict__ row_cnt,
                     const float* __restrict__ pred_program,
                     const int*   __restrict__ gt_program,
                     const int*   __restrict__ gt_prog_len,
                     const float* __restrict__ gt_intervals,
                     const float* __restrict__ pred_intervals,
                     const int*   __restrict__ gt_caps_count,
                     float* __restrict__ out) {
    const int tid = threadIdx.x;
    const float LOG2E = 1.4426950408889634f;
    const float LN2   = 0.6931471805599453f;

    float capN = 0.f, capC = 0.f, progN = 0.f, progC = 0.f, iouS = 0.f, ncapS = 0.f;

    // Caption partials (coalesced, 32 iters)
    for (int r = tid; r < NROWS; r += 256) {
        capN += row_nll[r];
        capC += row_cnt[r];
    }

    // Program CE: 4 rows per thread, 20-wide each
    for (int pr = tid; pr < NPROG; pr += 256) {
        const int b = pr >> 6, pos = pr & 63;
        if (pos < gt_prog_len[b]) {
            const float* __restrict__ rp = pred_program + (size_t)pr * PVC;
            float mx = -3.0e38f;
#pragma unroll
            for (int j = 0; j < PVC; ++j) mx = fmaxf(mx, rp[j]);
            float sm = 0.f;
#pragma unroll
            for (int j = 0; j < PVC; ++j) sm += hw_exp2((rp[j] - mx) * LOG2E);
            const int g = gt_program[pr];
            progN += mx + hw_log2(sm) * LN2 - rp[g];
            progC += 1.0f;
        }
    }

    // IoU: one (b, m) pair per thread
    {
        const int b = tid >> 4, mc = tid & 15;
        if (mc < gt_caps_count[b]) {
            const float* gi = gt_intervals + (size_t)(b * MC + mc) * 2;
            const float* pi = pred_intervals + (size_t)(b * MC + mc) * 2;
            float inter = fminf(pi[1], gi[1]) - fmaxf(pi[0], gi[0]);
            inter = fmaxf(inter, 0.0f);
            float uni = fmaxf(pi[1], gi[1]) - fminf(pi[0], gi[0]);
            iouS = inter / fmaxf(uni, 1e-8f);
        }
        if (tid < BSZ) ncapS = (float)gt_caps_count[tid];
    }

    // Stage the 6 quantities' 256 partials in LDS
    __shared__ float lds[6][256];
    lds[0][tid] = capN;  lds[1][tid] = capC;
    lds[2][tid] = progN; lds[3][tid] = progC;
    lds[4][tid] = iouS;  lds[5][tid] = ncapS;
    __syncthreads();

    // Wave 0 only (EXEC all-ones within the wave -> legal for WMMA).
    if (tid < 32) {
        const int lane = tid;
        v2f ones; ones.x = 1.0f; ones.y = 1.0f;   // B = 4x16 all-ones
        float tot[6];
#pragma unroll
        for (int q = 0; q < 6; ++q) {
            v8f c = {};
#pragma unroll
            for (int ch = 0; ch < 4; ++ch) {
                v2f a;                              // 64 partials per WMMA
                a.x = lds[q][ch * 64 + lane * 2];
                a.y = lds[q][ch * 64 + lane * 2 + 1];
                // D[m,n] = sum_k A[m,k] + C[m,n]  (chained accumulation)
                c = __builtin_amdgcn_wmma_f32_16x16x4_f32(
                        false, a, false, ones, (short)0, c, false, false);
            }
            // Column 0 of D: lane 0 holds M=0..7 across VGPRs, lane 16 holds M=8..15
            float tsum = c[0] + c[1] + c[2] + c[3] + c[4] + c[5] + c[6] + c[7];
            tsum += warp_xor_f(tsum, 16);
            tot[q] = tsum;
        }
        if (lane == 0) {
            const float cap_loss  = tot[0] / fmaxf(tot[1], 1.0f);
            const float prog_loss = tot[2] / fmaxf(tot[3], 1.0f);
            const float iou_loss  = 1.0f - tot[4] / fmaxf(tot[5], 1.0f);
            out[0] = cap_loss + prog_loss;
            out[1] = cap_loss;
            out[2] = prog_loss;
            out[3] = iou_loss;
        }
    }
}

extern "C" void kernel_launch(void* const* d_in, const int* in_sizes, int n_in,
                              void* d_out, int out_size, void* d_ws, size_t ws_size,
                              hipStream_t stream) {
    const int*   gt_captions    = (const int*)  d_in[0];
    const int*   gt_cap_lens    = (const int*)  d_in[1];
    const float* pred_captions  = (const float*)d_in[2];
    const int*   gt_program     = (const int*)  d_in[3];
    const int*   gt_prog_len    = (const int*)  d_in[4];
    const float* pred_program   = (const float*)d_in[5];
    const float* gt_intervals   = (const float*)d_in[6];
    const float* pred_intervals = (const float*)d_in[7];
    const int*   gt_caps_count  = (const int*)  d_in[8];
    // d_in[9] (pred_caps_count) unused by the reference math

    float* row_nll = (float*)d_ws;           // 8192 floats
    float* row_cnt = row_nll + NROWS;        // 8192 floats
    float* out     = (float*)d_out;          // 4 floats

    cap_ce_rows_kernel<<<NROWS, 256, 0, stream>>>(
        pred_captions, gt_captions, gt_cap_lens, gt_caps_count, row_nll, row_cnt);

    finalize_kernel<<<1, 256, 0, stream>>>(
        row_nll, row_cnt, pred_program, gt_program, gt_prog_len,
        gt_intervals, pred_intervals, gt_caps_count, out);
}